// WeightedLinearDynamics_47717086658700
// MI455X (gfx1250) — compile-verified
//
#include <hip/hip_runtime.h>
#include <hip/hip_bf16.h>

typedef float v2f __attribute__((ext_vector_type(2)));
typedef float v8f __attribute__((ext_vector_type(8)));

// D = A(16x4 f32) * B(4x16 f32) + C(16x16 f32)   -> v_wmma_f32_16x16x4_f32
__device__ __forceinline__ v8f wmma4(v2f a, v2f b, v8f c) {
  return __builtin_amdgcn_wmma_f32_16x16x4_f32(
      /*neg_a=*/false, a, /*neg_b=*/false, b,
      /*c_mod=*/(short)0, c, /*reuse_a=*/false, /*reuse_b=*/false);
}

__device__ __forceinline__ float fast_tanh(float x) {
#if __has_builtin(__builtin_amdgcn_tanhf)
  return __builtin_amdgcn_tanhf(x);   // native v_tanh_f32 on gfx1250
#else
  return tanhf(x);
#endif
}
__device__ __forceinline__ float sigm(float x) {
  return 0.5f + 0.5f * fast_tanh(0.5f * x);
}

// ---------------------------------------------------------------------------
// Kernel A: xg[s,b,g] = sum_n state[s,b,n] * W_ih[g,n] + b_ih[g] + b_hh[g]
// [8192,64] x [64,64]^T -> [8192,64]. Tiny; LDS-transposed weights (pad 65).
// ---------------------------------------------------------------------------
__global__ __launch_bounds__(256) void xg_kernel(
    const float* __restrict__ state, const float* __restrict__ W_ih,
    const float* __restrict__ b_ih,  const float* __restrict__ b_hh,
    float* __restrict__ xg) {
  __shared__ float xsh[256];
  __shared__ float wT[64 * 65];
  const int tid = threadIdx.x;
  xsh[tid] = state[(size_t)blockIdx.x * 256 + tid];
  for (int idx = tid; idx < 4096; idx += 256) {
    int g = idx >> 6, n = idx & 63;
    wT[n * 65 + g] = W_ih[idx];           // transpose, padded -> no bank conflicts
  }
  __syncthreads();
  const int r = tid >> 6, g = tid & 63;
  float acc = b_ih[g] + b_hh[g];
  const float* xr = &xsh[r * 64];
#pragma unroll 8
  for (int n = 0; n < 64; ++n) acc = fmaf(xr[n], wT[n * 65 + g], acc);
  xg[(size_t)blockIdx.x * 256 + tid] = acc;
}

// ---------------------------------------------------------------------------
// Kernel B: sequential LSTM scan, single workgroup (8 waves), WMMA recurrence.
//   gates = xg_t (as WMMA C operand) + h[64,16] @ W_hh^T[16,64]
// Wave w: rows b0 = (w&3)*16, cols jbase = (w>>2)*32 (two 16-col tiles).
// xg C-fragments double-buffered one step ahead to hide global latency.
// ---------------------------------------------------------------------------
__global__ __launch_bounds__(256) void lstm_kernel(
    const float* __restrict__ xg,    // [128][64][64]
    const float* __restrict__ W_hh,  // [64][16]
    float* __restrict__ wout) {      // [128][64][16]
  __shared__ float h_lds[64 * 16];
  __shared__ float gate[64 * 64];

  const int lane = threadIdx.x;            // 0..31
  const int wv   = threadIdx.y;            // 0..7
  const int tid  = wv * 32 + lane;
  const int half = lane >> 4;
  const int l15  = lane & 15;
  const int b0    = (wv & 3) * 16;
  const int jbase = (wv >> 2) * 32;

  // B fragments (W_hh^T tiles): constant over all 128 steps, keep in VGPRs.
  v2f bf[2][4];
#pragma unroll
  for (int ct = 0; ct < 2; ++ct) {
    const int j = jbase + ct * 16 + l15;
#pragma unroll
    for (int q = 0; q < 4; ++q) {
      bf[ct][q].x = W_hh[j * 16 + 4 * q + 2 * half + 0];
      bf[ct][q].y = W_hh[j * 16 + 4 * q + 2 * half + 1];
    }
  }

  // elementwise mapping: thread -> hidden k, 4 batch rows
  const int ek  = tid & 15;
  const int eb0 = (tid >> 4) * 4;
  float cst[4] = {0.f, 0.f, 0.f, 0.f};

  for (int i = tid; i < 64 * 16; i += 256) h_lds[i] = 0.f;
  __syncthreads();

  auto loadC = [&](int t, int ct) -> v8f {   // xg tile as C/D-layout fragment
    v8f c;
    const float* base = xg + (size_t)t * 4096 + jbase + ct * 16 + l15;
#pragma unroll
    for (int r = 0; r < 8; ++r) c[r] = base[(size_t)(b0 + r + 8 * half) * 64];
    return c;
  };

  auto step = [&](int t, v8f cf0, v8f cf1) {
    v2f af[4];                               // A fragments from h_lds
#pragma unroll
    for (int q = 0; q < 4; ++q) {
      const int br = (b0 + l15) * 16;
      af[q].x = h_lds[br + 4 * q + 2 * half + 0];
      af[q].y = h_lds[br + 4 * q + 2 * half + 1];
    }
    v8f d0 = cf0, d1 = cf1;
#pragma unroll
    for (int q = 0; q < 4; ++q) d0 = wmma4(af[q], bf[0][q], d0);
#pragma unroll
    for (int q = 0; q < 4; ++q) d1 = wmma4(af[q], bf[1][q], d1);
#pragma unroll
    for (int r = 0; r < 8; ++r) {
      const int brow = (b0 + r + 8 * half) * 64;
      gate[brow + jbase +  0 + l15] = d0[r];
      gate[brow + jbase + 16 + l15] = d1[r];
    }
    __syncthreads();
#pragma unroll
    for (int u = 0; u < 4; ++u) {            // gate order i,f,g,o
      const int b = eb0 + u;
      const float iv = sigm(gate[b * 64 +  0 + ek]);
      const float fv = sigm(gate[b * 64 + 16 + ek]);
      const float gv = fast_tanh(gate[b * 64 + 32 + ek]);
      const float ov = sigm(gate[b * 64 + 48 + ek]);
      cst[u] = fv * cst[u] + iv * gv;
      const float hv = ov * fast_tanh(cst[u]);
      h_lds[b * 16 + ek] = hv;
      wout[((size_t)t * 64 + b) * 16 + ek] = hv;
    }
    __syncthreads();
  };

  v8f ca0 = loadC(0, 0), ca1 = loadC(0, 1);
  for (int t = 0; t < 128; t += 2) {
    v8f cb0 = loadC(t + 1, 0), cb1 = loadC(t + 1, 1);  // prefetch t+1
    step(t, ca0, ca1);
    if (t + 2 < 128) { ca0 = loadC(t + 2, 0); ca1 = loadC(t + 2, 1); }
    step(t + 1, cb0, cb1);
  }
}

// ---------------------------------------------------------------------------
// Kernel C: out[8192, ncols] = w[8192,16] @ Mat[16, ncols].
// Block = 8 waves. 16x64 slab of Mat staged in LDS once; each wave: one
// 16-row tile, 4 column tiles of 4 chained v_wmma_f32_16x16x4_f32.
// 302 MB total output -> nontemporal streaming stores.
// ---------------------------------------------------------------------------
__global__ __launch_bounds__(256) void expand_kernel(
    const float* __restrict__ w,    // [8192][16]
    const float* __restrict__ Mat,  // [16][ncols]
    float* __restrict__ out,        // [8192][ncols]
    int ncols) {
  __shared__ float tile[16 * 64];
  const int lane = threadIdx.x, wv = threadIdx.y;
  const int tid = wv * 32 + lane;
  const int jg0 = blockIdx.x * 64;

  {
    const int k = tid >> 4;
    const int cc = (tid & 15) << 2;
    const float4 v =
        *reinterpret_cast<const float4*>(Mat + (size_t)k * ncols + jg0 + cc);
    *reinterpret_cast<float4*>(&tile[k * 64 + cc]) = v;
  }
  __syncthreads();

  const int half = lane >> 4, l15 = lane & 15;
  const int sb0 = (blockIdx.y * 8 + wv) * 16;

  const float4* wr = reinterpret_cast<const float4*>(w + (size_t)(sb0 + l15) * 16);
  v2f af[4];
#pragma unroll
  for (int q = 0; q < 4; ++q) {
    const float4 wq = wr[q];
    af[q].x = half ? wq.z : wq.x;
    af[q].y = half ? wq.w : wq.y;
  }

#pragma unroll
  for (int ct = 0; ct < 4; ++ct) {
    const int c0 = ct * 16;
    v8f d = {};
#pragma unroll
    for (int q = 0; q < 4; ++q) {
      v2f bq;
      bq.x = tile[(4 * q + 2 * half + 0) * 64 + c0 + l15];
      bq.y = tile[(4 * q + 2 * half + 1) * 64 + c0 + l15];
      d = wmma4(af[q], bq, d);
    }
    float* obase = out + (size_t)(sb0 + 8 * half) * ncols + jg0 + c0 + l15;
#pragma unroll
    for (int r = 0; r < 8; ++r)
      __builtin_nontemporal_store(d[r], obase + (size_t)r * ncols);
  }
}

// ---------------------------------------------------------------------------
extern "C" void kernel_launch(void* const* d_in, const int* in_sizes, int n_in,
                              void* d_out, int out_size, void* d_ws,
                              size_t ws_size, hipStream_t stream) {
  const float* state = (const float*)d_in[0];  // [128,64,64]
  const float* W_ih  = (const float*)d_in[1];  // [64,64]
  const float* W_hh  = (const float*)d_in[2];  // [64,16]
  const float* b_ih  = (const float*)d_in[3];  // [64]
  const float* b_hh  = (const float*)d_in[4];  // [64]
  const float* As    = (const float*)d_in[5];  // [16,64,64]
  const float* Bs    = (const float*)d_in[6];  // [16,64,32]
  const float* Cs    = (const float*)d_in[7];  // [16,48,64]

  float* xg   = (float*)d_ws;                  // 128*64*64   = 524288 f
  float* wbuf = xg + (size_t)128 * 64 * 64;    // 128*64*16   = 131072 f

  xg_kernel<<<2048, 256, 0, stream>>>(state, W_ih, b_ih, b_hh, xg);
  lstm_kernel<<<1, dim3(32, 8), 0, stream>>>(xg, W_hh, wbuf);

  float* outA = (float*)d_out;                       // [8192,4096]
  float* outB = outA + (size_t)8192 * 4096;          // [8192,2048]
  float* outC = outB + (size_t)8192 * 2048;          // [8192,3072]
  expand_kernel<<<dim3(64, 64), dim3(32, 8), 0, stream>>>(wbuf, As, outA, 4096);
  expand_kernel<<<dim3(32, 64), dim3(32, 8), 0, stream>>>(wbuf, Bs, outB, 2048);
  expand_kernel<<<dim3(48, 64), dim3(32, 8), 0, stream>>>(wbuf, Cs, outC, 3072);
}